// GraphSAGE_31104153158139
// MI455X (gfx1250) — compile-verified
//
#include <hip/hip_runtime.h>
#include <hip/hip_bf16.h>

typedef __attribute__((ext_vector_type(2))) float v2f;
typedef __attribute__((ext_vector_type(8))) float v8f;

#define DF 128
#define LDSPITCH 132   // 128 + 4 pad: (4*ln + k) % 64 -> conflict-free ds_load_b64

// ---------------- utility kernels ----------------

__global__ void zero_f32(float* __restrict__ p, size_t n) {
    size_t i = (size_t)blockIdx.x * blockDim.x + threadIdx.x;
    size_t stride = (size_t)gridDim.x * blockDim.x;
    for (; i < n; i += stride) p[i] = 0.0f;
}

__global__ void degree_kernel(const long long* __restrict__ dst,
                              float* __restrict__ deg, int E) {
    int e = blockIdx.x * blockDim.x + threadIdx.x;
    if (e < E) atomicAdd(&deg[dst[e]], 1.0f);
}

// One wave32 per edge; each lane moves 4 floats (float4 load, 4 f32 atomics).
__global__ void scatter_kernel(const float* __restrict__ feat,
                               const long long* __restrict__ src,
                               const long long* __restrict__ dstI,
                               float* __restrict__ agg, int E) {
    long long t = (long long)blockIdx.x * blockDim.x + threadIdx.x;
    long long e = t >> 5;
    int lane = (int)(t & 31);
    if (e >= E) return;
    long long s = src[e];
    long long d = dstI[e];
    float4 v = *reinterpret_cast<const float4*>(feat + s * DF + lane * 4);
    float* dp = agg + d * DF + lane * 4;
    atomicAdd(dp + 0, v.x);
    atomicAdd(dp + 1, v.y);
    atomicAdd(dp + 2, v.z);
    atomicAdd(dp + 3, v.w);
}

// ---------------- fused SAGE layer: out = relu?(mean@Wl^T + x@Wr^T + b) ----------------
// Block = 256 threads = 8 waves. Block handles 16 node rows; wave w handles
// output columns [16w, 16w+16). Requires N % 16 == 0 (true: 100000/16 = 6250).
__global__ __launch_bounds__(256) void sage_gemm(
    const float* __restrict__ agg, const float* __restrict__ deg,
    const float* __restrict__ xin,
    const float* __restrict__ Wl, const float* __restrict__ Wr,
    const float* __restrict__ bias,
    float* __restrict__ out, int do_relu)
{
    __shared__ float meanT[16][LDSPITCH];
    __shared__ float selfT[16][LDSPITCH];

    const int tid = threadIdx.x;
    const size_t row0 = (size_t)blockIdx.x * 16;

    // Stage 16x128 mean tile (agg/deg fused) and 16x128 self tile into LDS.
    // 2048 floats per tile / 256 threads = 8 floats (2x float4) per thread.
    {
        int idx = tid * 8;
        int r = idx >> 7;
        int c = idx & 127;
        size_t g = row0 + r;
        float invd = 1.0f / fmaxf(deg[g], 1.0f);
        const float4* ap = reinterpret_cast<const float4*>(agg + g * DF + c);
        float4 a0 = ap[0], a1 = ap[1];
        a0.x *= invd; a0.y *= invd; a0.z *= invd; a0.w *= invd;
        a1.x *= invd; a1.y *= invd; a1.z *= invd; a1.w *= invd;
        *reinterpret_cast<float4*>(&meanT[r][c])     = a0;
        *reinterpret_cast<float4*>(&meanT[r][c + 4]) = a1;
        const float4* xp = reinterpret_cast<const float4*>(xin + g * DF + c);
        *reinterpret_cast<float4*>(&selfT[r][c])     = xp[0];
        *reinterpret_cast<float4*>(&selfT[r][c + 4]) = xp[1];
    }
    __syncthreads();

    const int wave = tid >> 5;
    const int lane = tid & 31;
    const int ln   = lane & 15;   // row (A) / col (B,C,D) within tile
    const int hi   = lane >> 4;   // K-half select for A/B; M+8 select for C/D
    const int j0   = wave * 16;   // output column base

    // C/D layout: VGPR r, lanes 0-15 -> M=r, lanes 16-31 -> M=r+8; N = lane%16.
    v8f c;
    float bj = bias[j0 + ln];
    for (int r = 0; r < 8; ++r) c[r] = bj;

    const float* wlRow = Wl + (size_t)(j0 + ln) * DF;  // B[k][j] = Wl[j][k]
    const float* wrRow = Wr + (size_t)(j0 + ln) * DF;

    for (int kk = 0; kk < DF; kk += 4) {
        const int ka = kk + 2 * hi;   // lanes 0-15: K=kk,kk+1; lanes 16-31: K=kk+2,kk+3
        v2f a, b;
        // Wl-term: A = mean tile
        a = *reinterpret_cast<const v2f*>(&meanT[ln][ka]);
        b = *reinterpret_cast<const v2f*>(wlRow + ka);
        c = __builtin_amdgcn_wmma_f32_16x16x4_f32(false, a, false, b,
                                                  (short)0, c, false, false);
        // Wr-term: A = self tile
        a = *reinterpret_cast<const v2f*>(&selfT[ln][ka]);
        b = *reinterpret_cast<const v2f*>(wrRow + ka);
        c = __builtin_amdgcn_wmma_f32_16x16x4_f32(false, a, false, b,
                                                  (short)0, c, false, false);
    }

    for (int r = 0; r < 8; ++r) {
        float v = c[r];
        if (do_relu) v = fmaxf(v, 0.0f);
        out[(row0 + (size_t)(r + 8 * hi)) * DF + j0 + ln] = v;
    }
}

// ---------------- launcher ----------------

extern "C" void kernel_launch(void* const* d_in, const int* in_sizes, int n_in,
                              void* d_out, int out_size, void* d_ws, size_t ws_size,
                              hipStream_t stream) {
    const float*     x    = (const float*)d_in[0];
    const long long* ei   = (const long long*)d_in[1];
    const float*     Wl1  = (const float*)d_in[2];
    const float*     Wr1  = (const float*)d_in[3];
    const float*     b1   = (const float*)d_in[4];
    const float*     Wl2  = (const float*)d_in[5];
    const float*     Wr2  = (const float*)d_in[6];
    const float*     b2   = (const float*)d_in[7];
    float* out = (float*)d_out;

    const int N = in_sizes[0] / DF;   // 100000
    const int E = in_sizes[1] / 2;    // 1600000
    const long long* src = ei;        // edge_index[0]
    const long long* dst = ei + E;    // edge_index[1]

    // Workspace layout (needs ~103 MB): deg[N] | agg[N*128] | h[N*128]
    float* deg = (float*)d_ws;
    float* agg = deg + N;                       // N*4 B offset (16B aligned)
    float* h   = agg + (size_t)N * DF;

    const int scatterBlocks = (int)(((long long)E * 32 + 255) / 256);
    const int degBlocks     = (E + 255) / 256;
    const int gemmBlocks    = N / 16;

    // Layer 1
    zero_f32<<<2048, 256, 0, stream>>>(deg, (size_t)N + (size_t)N * DF); // deg+agg contiguous
    degree_kernel<<<degBlocks, 256, 0, stream>>>(dst, deg, E);
    scatter_kernel<<<scatterBlocks, 256, 0, stream>>>(x, src, dst, agg, E);
    sage_gemm<<<gemmBlocks, 256, 0, stream>>>(agg, deg, x, Wl1, Wr1, b1, h, 1);

    // Layer 2 (degree is identical, only re-zero agg)
    zero_f32<<<2048, 256, 0, stream>>>(agg, (size_t)N * DF);
    scatter_kernel<<<scatterBlocks, 256, 0, stream>>>(h, src, dst, agg, E);
    sage_gemm<<<gemmBlocks, 256, 0, stream>>>(agg, deg, h, Wl2, Wr2, b2, out, 0);
}